// SparseGlobalPool_8358006358531
// MI455X (gfx1250) — compile-verified
//
#include <hip/hip_runtime.h>

typedef float v2f __attribute__((ext_vector_type(2)));
typedef float v8f __attribute__((ext_vector_type(8)));

#define TPB 256        // 8 wave32s per block
#define NBLK 512       // 2048 point-subsets, 4096 waves in flight
#define NSEG 32
#define NCH 128

// ---------------- kernel 0: zero output sums + counts ----------------
__global__ __launch_bounds__(TPB) void seg_zero(float* __restrict__ out,
                                                float* __restrict__ cnt) {
  int i = blockIdx.x * TPB + threadIdx.x;
  if (i < NSEG * NCH) out[i] = 0.0f;
  if (i < NSEG) cnt[i] = 0.0f;
}

// ---------------- kernel 1: WMMA one-hot segment reduction ----------------
__global__ __launch_bounds__(TPB) void seg_pool(const float* __restrict__ feat,
                                                const int* __restrict__ bids,
                                                float* __restrict__ out_sum,
                                                float* __restrict__ out_cnt,
                                                int N) {
  __shared__ float lds_acc[NSEG * NCH];
  __shared__ float lds_cnt[NSEG];

  const int tid = threadIdx.x;
  for (int i = tid; i < NSEG * NCH; i += TPB) lds_acc[i] = 0.0f;
  if (tid < NSEG) lds_cnt[tid] = 0.0f;
  __syncthreads();

  // Wave id is wave-uniform: force it scalar so p / base pointers / loop
  // control live in SGPRs (saddr-form loads, s_cbranch loop, EXEC untouched).
  const int waveU = __builtin_amdgcn_readfirstlane(tid >> 5);  // 0..7
  const int lane = tid & 31;
  const int half = lane >> 4;    // lanes 0-15 vs 16-31
  const int m    = lane & 15;    // row/col within a 16-tile
  const int q    = waveU & 1;    // channel half: 64 channels per wave
  const int pr   = waveU >> 1;   // point-subset within block (0..3)
  const int chBase = q * 64;

  const long long pairId = (long long)blockIdx.x * 4 + pr;
  const long long nPairs = (long long)gridDim.x * 4;
  const long long step   = nPairs * 4;

  // C/D accumulators: [segment half][channel tile] -> 16x16 f32 = 8 VGPRs each
  v8f acc[2][4] = {};
  // point counts via one-hot row sums (dup across halves & tiles; merged once)
  float cnt_acc[2] = {0.0f, 0.0f};

  // A (16x4): lanes 0-15 hold K=0,1; lanes 16-31 hold K=2,3.
  // B (4x16): VGPR0 = rows {0,2}, VGPR1 = rows {1,3}, striped identically.
  const int k0 = half * 2;                 // per-lane
  const int laneOff = k0 * NCH + m;        // loop-invariant per-lane voffset

  long long p = pairId * 4;
  for (; p + 3 < N; p += step) {
    const int2 bb = *(const int2*)(bids + p + k0);       // saddr + small voffset
    const float* base = feat + p * NCH + chBase;         // scalar base (s_add/iter)

    v2f bT[4];
#pragma unroll
    for (int cg = 0; cg < 4; ++cg) {                     // channels chBase+cg*16+m
      bT[cg].x = __builtin_nontemporal_load(base + laneOff + cg * 16);
      bT[cg].y = __builtin_nontemporal_load(base + laneOff + NCH + cg * 16);
    }

#pragma unroll
    for (int sg = 0; sg < 2; ++sg) {
      const int seg = sg * 16 + m;
      v2f a;
      a.x = (bb.x == seg) ? 1.0f : 0.0f;
      a.y = (bb.y == seg) ? 1.0f : 0.0f;
      cnt_acc[sg] += a.x + a.y;                          // branch-free counting
#pragma unroll
      for (int cg = 0; cg < 4; ++cg)
        acc[sg][cg] = __builtin_amdgcn_wmma_f32_16x16x4_f32(
            false, a, false, bT[cg], (short)0, acc[sg][cg], false, false);
    }
  }

  // Guarded tail (never taken when N % 4 == 0; scalar-uniform branch)
  if (p < N) {
    const long long i0 = p + k0;
    const long long i1 = i0 + 1;
    const int b0 = (i0 < N) ? bids[i0] : -1;
    const int b1 = (i1 < N) ? bids[i1] : -1;
    const float* r0 = feat + (i0 < N ? i0 : 0) * NCH + chBase + m;
    const float* r1 = feat + (i1 < N ? i1 : 0) * NCH + chBase + m;
    v2f bT[4];
#pragma unroll
    for (int cg = 0; cg < 4; ++cg) {
      bT[cg].x = r0[cg * 16];
      bT[cg].y = r1[cg * 16];
    }
#pragma unroll
    for (int sg = 0; sg < 2; ++sg) {
      const int seg = sg * 16 + m;
      v2f a;
      a.x = (b0 == seg) ? 1.0f : 0.0f;
      a.y = (b1 == seg) ? 1.0f : 0.0f;
      cnt_acc[sg] += a.x + a.y;
#pragma unroll
      for (int cg = 0; cg < 4; ++cg)
        acc[sg][cg] = __builtin_amdgcn_wmma_f32_16x16x4_f32(
            false, a, false, bT[cg], (short)0, acc[sg][cg], false, false);
    }
  }

  // Register tiles -> block LDS accumulator.
  // C/D layout: VGPR j holds M=j (lanes 0-15) / M=j+8 (lanes 16-31), N=m.
#pragma unroll
  for (int sg = 0; sg < 2; ++sg)
#pragma unroll
    for (int cg = 0; cg < 4; ++cg)
#pragma unroll
      for (int j = 0; j < 8; ++j) {
        const int seg = sg * 16 + j + 8 * half;
        const int ch  = chBase + cg * 16 + m;
        atomicAdd(&lds_acc[seg * NCH + ch], acc[sg][cg][j]);
      }
  // one counting wave per point-subset (q==0); halves hold disjoint points
  if (q == 0) {
#pragma unroll
    for (int sg = 0; sg < 2; ++sg)
      atomicAdd(&lds_cnt[sg * 16 + m], cnt_acc[sg]);
  }
  __syncthreads();

  // Block partials -> global (L2-resident 16.5 KB region).
  for (int i = tid; i < NSEG * NCH; i += TPB)
    atomicAdd(&out_sum[i], lds_acc[i]);
  if (tid < NSEG) atomicAdd(&out_cnt[tid], lds_cnt[tid]);
}

// ---------------- kernel 2: in-place divide, zero empty segments ----------------
__global__ __launch_bounds__(TPB) void seg_final(float* __restrict__ out,
                                                 const float* __restrict__ cnt) {
  int i = blockIdx.x * TPB + threadIdx.x;
  if (i < NSEG * NCH) {
    float c = cnt[i >> 7];        // i / NCH
    float s = out[i];
    out[i] = (c > 0.0f) ? (s / c) : 0.0f;
  }
}

extern "C" void kernel_launch(void* const* d_in, const int* in_sizes, int n_in,
                              void* d_out, int out_size, void* d_ws, size_t ws_size,
                              hipStream_t stream) {
  const float* feat = (const float*)d_in[0];
  const int*   bids = (const int*)d_in[1];   // integer input -> const int*
  float* out = (float*)d_out;                // 32*128 segment sums -> means
  float* cnt = (float*)d_ws;                 // 32 floats of scratch
  const int N = in_sizes[0] / NCH;

  seg_zero <<<(NSEG * NCH + TPB - 1) / TPB, TPB, 0, stream>>>(out, cnt);
  seg_pool <<<NBLK, TPB, 0, stream>>>(feat, bids, out, cnt, N);
  seg_final<<<(NSEG * NCH + TPB - 1) / TPB, TPB, 0, stream>>>(out, cnt);
}